// RepeatDecoder_90220083019795
// MI455X (gfx1250) — compile-verified
//
#include <hip/hip_runtime.h>
#include <math.h>

#define VOCAB_N 100000
#define PAD_ID (VOCAB_N - 2)
#define INTEREST_ID (VOCAB_N - 1)
#define SEQ 200
#define HID 128

typedef __attribute__((ext_vector_type(2))) float v2f;
typedef __attribute__((ext_vector_type(8))) float v8f;

// Hardware tanh (gfx1250 TRANS op). Prefer the clang builtin; otherwise emit
// v_tanh_f32 directly (v_nop after the trans op per the co-execution hazard
// rule, since the compiler cannot schedule around inline asm).
__device__ __forceinline__ float htanh(float x) {
#if defined(__has_builtin)
#if __has_builtin(__builtin_amdgcn_tanhf)
    return __builtin_amdgcn_tanhf(x);
#else
    float r;
    asm volatile("v_tanh_f32 %0, %1\n\tv_nop" : "=v"(r) : "v"(x));
    return r;
#endif
#else
    return tanhf(x);
#endif
}

// ---------------- Kernel 1: zero the (1024, 100000) f32 output ----------------
__global__ __launch_bounds__(256) void zero_f4(float4* __restrict__ out, long long n4) {
    long long i = (long long)blockIdx.x * blockDim.x + threadIdx.x;
    if (i < n4) out[i] = make_float4(0.f, 0.f, 0.f, 0.f);
}

// ---------------- Kernel 2: scores via f32 WMMA + softmax + scatter ----------------
// One block per batch row b. 256 threads = 8 wave32s.
// scores[b,s] = dot(tanh(hidden[b,s,:] + hidden[b,0,:]), w) + bias
__global__ __launch_bounds__(256) void repeat_decoder(
    const float* __restrict__ hs,   // (1024, 200, 128) f32
    const int*   __restrict__ ids,  // (1024, 200) int
    const float* __restrict__ w,    // (128,)
    const float* __restrict__ bp,   // (1,)
    float*       __restrict__ out)  // (1024, 100000)
{
    __shared__ float keysLds[HID];
    __shared__ float wLds[HID];
    __shared__ float scoresLds[13 * 16]; // 208, last 8 unused
    __shared__ float red[256];

    const int b    = blockIdx.x;
    const int t    = threadIdx.x;
    const int lane = t & 31;
    const int wv   = t >> 5;

    const float* hrow = hs + (size_t)b * SEQ * HID;

    if (t < HID) {
        keysLds[t] = hrow[t];   // hidden[b, 0, t]
        wLds[t]    = w[t];
    }
    __syncthreads();

    // 13 tiles of 16 sequence positions, round-robined over the 8 waves.
    for (int tile = wv; tile < 13; tile += 8) {
        const int s0 = tile * 16;
        int srow = s0 + (lane & 15);
        if (srow > SEQ - 1) srow = SEQ - 1;      // clamp: garbage rows are never consumed
        const int cbase = (lane >> 4) << 1;      // lanes 0-15 -> K=0,1 ; lanes 16-31 -> K=2,3
        const float* rp = hrow + srow * HID;

        v8f acc = {0.f, 0.f, 0.f, 0.f, 0.f, 0.f, 0.f, 0.f};
        #pragma unroll
        for (int k = 0; k < 32; ++k) {           // K = 128 in chunks of 4
            const int col = (k << 2) + cbase;
            float2 x = *reinterpret_cast<const float2*>(rp + col);
            v2f a, bf;
            a.x  = htanh(x.x + keysLds[col]);
            a.y  = htanh(x.y + keysLds[col + 1]);
            bf.x = wLds[col];                    // B[k][n] = w[k] broadcast over N
            bf.y = wLds[col + 1];
            // D = A(16x4,f32) * B(4x16,f32) + C ; all 16 columns of D equal the score
            acc = __builtin_amdgcn_wmma_f32_16x16x4_f32(
                      false, a, false, bf, (short)0, acc, false, false);
        }
        // D layout: VGPR j -> M=j (lanes 0-15), M=8+j (lanes 16-31); take column N=0.
        if (lane == 0) {
            #pragma unroll
            for (int j = 0; j < 8; ++j) scoresLds[s0 + j] = acc[j];
        } else if (lane == 16) {
            #pragma unroll
            for (int j = 0; j < 8; ++j) scoresLds[s0 + 8 + j] = acc[j];
        }
    }
    __syncthreads();

    // -------- masked softmax over S=200 --------
    const float bias = bp[0];
    int   myid = 0;
    float sc   = -INFINITY;
    if (t < SEQ) {
        myid = ids[(size_t)b * SEQ + t];
        const bool m = (myid != PAD_ID) && (myid != INTEREST_ID);
        sc = m ? (scoresLds[t] + bias) : -INFINITY;
    }

    // block max
    red[t] = sc;
    __syncthreads();
    #pragma unroll
    for (int off = 128; off > 0; off >>= 1) {
        if (t < off) red[t] = fmaxf(red[t], red[t + off]);
        __syncthreads();
    }
    const float mx = red[0];
    __syncthreads();

    // block sum of exp
    const float e = (t < SEQ) ? expf(sc - mx) : 0.f;
    red[t] = e;
    __syncthreads();
    #pragma unroll
    for (int off = 128; off > 0; off >>= 1) {
        if (t < off) red[t] += red[t + off];
        __syncthreads();
    }
    const float sum = red[0];

    // scatter-add the attention weights; block owns this output row, atomics
    // only resolve duplicate token ids within the row.
    if (t < SEQ) {
        atomicAdd(&out[(size_t)b * VOCAB_N + myid], e / sum);
    }
}

extern "C" void kernel_launch(void* const* d_in, const int* in_sizes, int n_in,
                              void* d_out, int out_size, void* d_ws, size_t ws_size,
                              hipStream_t stream) {
    const float* hs  = (const float*)d_in[0]; // hidden_states (1024,200,128)
    const int*   ids = (const int*)  d_in[1]; // input_ids (1024,200)
    const float* w   = (const float*)d_in[2]; // w_proj (128,)
    const float* bp  = (const float*)d_in[3]; // b_proj (1,)
    float*       out = (float*)d_out;         // (1024, 100000)

    // zero the output (409.6 MB) with wide stores
    const long long n4 = (long long)out_size / 4;
    const int zthreads = 256;
    const long long zblocks = (n4 + zthreads - 1) / zthreads;
    zero_f4<<<(unsigned)zblocks, zthreads, 0, stream>>>((float4*)out, n4);

    // main fused kernel: one block per batch row
    repeat_decoder<<<1024, 256, 0, stream>>>(hs, ids, w, bp, out);
}